// lstm_model_55602646614621
// MI455X (gfx1250) — compile-verified
//
#include <hip/hip_runtime.h>
#include <hip/hip_bf16.h>

typedef __attribute__((ext_vector_type(2))) float v2f;
typedef __attribute__((ext_vector_type(8))) float v8f;
typedef __attribute__((ext_vector_type(4))) float f32x4;

#define BATCH 512
#define TSTEPS 20
#define HID 1024
#define G4 4096
#define TM 128
#define TN 128
#define KC 16
#define LDS_STRIDE 20   // floats: rows 16B-aligned, conflict-free v2f frag reads

// ---------------------------------------------------------------------------
// Tiled f32 WMMA GEMM:  C[r][j] = bias[j] + sum_k A1[r][k]*W1w[j][k]
//                                         + sum_k A2[r][k]*W2w[j][k]  (optional)
// A is [M][K] row-major; W is the weight in ORIGINAL [N][K] row-major layout
// (B-tile is staged n-major in LDS, so no pre-transpose needed).
// Block = 256 threads (8 waves); 128x128 C tile; wave computes 32x64.
// Register-prefetch + double-buffered LDS, one barrier per K-chunk.
// ---------------------------------------------------------------------------
struct TileRegs { f32x4 a0, a1, b0, b1; };

__device__ __forceinline__ void load_tile(const float* __restrict__ A,
                                          const float* __restrict__ W,
                                          int K, int row0, int col0, int k0,
                                          int tid, TileRegs& t)
{
    const int r  = tid >> 2;            // 0..63
    const int kc = (tid & 3) << 2;      // 0,4,8,12
    t.a0 = *(const f32x4*)(A + (size_t)(row0 + r)      * K + k0 + kc);
    t.a1 = *(const f32x4*)(A + (size_t)(row0 + r + 64) * K + k0 + kc);
    t.b0 = *(const f32x4*)(W + (size_t)(col0 + r)      * K + k0 + kc);
    t.b1 = *(const f32x4*)(W + (size_t)(col0 + r + 64) * K + k0 + kc);
}

__device__ __forceinline__ void store_tile(float* __restrict__ As, float* __restrict__ Bs,
                                           int tid, const TileRegs& t)
{
    const int r  = tid >> 2;
    const int kc = (tid & 3) << 2;
    *(f32x4*)(&As[(r)      * LDS_STRIDE + kc]) = t.a0;
    *(f32x4*)(&As[(r + 64) * LDS_STRIDE + kc]) = t.a1;
    *(f32x4*)(&Bs[(r)      * LDS_STRIDE + kc]) = t.b0;
    *(f32x4*)(&Bs[(r + 64) * LDS_STRIDE + kc]) = t.b1;
}

__launch_bounds__(256)
__global__ void gemm_wmma_f32(const float* __restrict__ A1, const float* __restrict__ W1w, int K1,
                              const float* __restrict__ A2, const float* __restrict__ W2w, int K2,
                              const float* __restrict__ bias, float* __restrict__ C,
                              int N, int relu)
{
    __shared__ float AsBuf[2][TM * LDS_STRIDE];
    __shared__ float BsBuf[2][TN * LDS_STRIDE];

    const int tid  = threadIdx.x;
    const int lane = tid & 31;
    const int wid  = tid >> 5;
    const int wm   = wid & 3;        // 4 waves along M
    const int wn   = wid >> 2;       // 2 waves along N
    const int m16  = lane & 15;
    const int hi   = lane >> 4;
    const int row0 = blockIdx.y * TM;
    const int col0 = blockIdx.x * TN;

    v8f acc[2][4] = {};

    const int nc1 = K1 / KC;
    const int nc2 = (A2 != nullptr) ? (K2 / KC) : 0;
    const int nc  = nc1 + nc2;

    // chunk c -> (A, W, K, k0)
    auto chunk_src = [&](int c, const float*& A, const float*& W, int& K, int& k0) {
        if (c < nc1) { A = A1; W = W1w; K = K1; k0 = c * KC; }
        else         { A = A2; W = W2w; K = K2; k0 = (c - nc1) * KC; }
    };

    // Prologue: stage chunk 0 into buffer 0.
    {
        const float *A, *W; int K, k0;
        chunk_src(0, A, W, K, k0);
        TileRegs t;
        load_tile(A, W, K, row0, col0, k0, tid, t);
        store_tile(AsBuf[0], BsBuf[0], tid, t);
    }

    for (int c = 0; c < nc; ++c) {
        __syncthreads();

        // Prefetch next chunk global->regs while WMMAs run on current buffer.
        TileRegs tn;
        if (c + 1 < nc) {
            const float *A, *W; int K, k0;
            chunk_src(c + 1, A, W, K, k0);
            load_tile(A, W, K, row0, col0, k0, tid, tn);
        }

        const float* __restrict__ As = AsBuf[c & 1];
        const float* __restrict__ Bs = BsBuf[c & 1];

        #pragma unroll
        for (int kk = 0; kk < 4; ++kk) {
            const int kb = kk * 4 + 2 * hi;
            v2f a[2];
            v2f b[4];
            #pragma unroll
            for (int i = 0; i < 2; ++i)
                a[i] = *(const v2f*)(&As[(wm * 32 + i * 16 + m16) * LDS_STRIDE + kb]);
            #pragma unroll
            for (int j = 0; j < 4; ++j)
                b[j] = *(const v2f*)(&Bs[(wn * 64 + j * 16 + m16) * LDS_STRIDE + kb]);
            #pragma unroll
            for (int i = 0; i < 2; ++i)
                #pragma unroll
                for (int j = 0; j < 4; ++j)
                    acc[i][j] = __builtin_amdgcn_wmma_f32_16x16x4_f32(
                        false, a[i], false, b[j], (short)0, acc[i][j], false, false);
        }

        if (c + 1 < nc)
            store_tile(AsBuf[(c + 1) & 1], BsBuf[(c + 1) & 1], tid, tn);
    }

    // Epilogue: bias + optional ReLU, direct f32 stores.
    #pragma unroll
    for (int i = 0; i < 2; ++i) {
        #pragma unroll
        for (int j = 0; j < 4; ++j) {
            const int col = col0 + wn * 64 + j * 16 + m16;
            const float bv = bias ? bias[col] : 0.0f;
            #pragma unroll
            for (int v = 0; v < 8; ++v) {
                const int row = row0 + wm * 32 + i * 16 + v + 8 * hi;
                float val = acc[i][j][v] + bv;
                if (relu) val = fmaxf(val, 0.0f);
                C[(size_t)row * N + col] = val;
            }
        }
    }
}

// ---------------------------------------------------------------------------
// Build X[(t*B + b)][1024] = concat(rel_emb[rel[b][t]], ent_emb[ent[b][t]])
// ---------------------------------------------------------------------------
__global__ void build_x_kernel(const int* __restrict__ ent, const int* __restrict__ rel,
                               const float* __restrict__ ent_emb, const float* __restrict__ rel_emb,
                               float* __restrict__ X)
{
    const int r = blockIdx.x;          // r = t*B + b
    const int b = r & (BATCH - 1);
    const int t = r >> 9;
    const int tid = threadIdx.x;       // 256 threads * 4 floats = 1024
    const int ridx = rel[b * TSTEPS + t];
    const int eidx = ent[b * TSTEPS + t];
    f32x4 v;
    if (tid < 128) v = *(const f32x4*)(rel_emb + (size_t)ridx * 512 + tid * 4);
    else           v = *(const f32x4*)(ent_emb + (size_t)eidx * 512 + (tid - 128) * 4);
    *(f32x4*)(X + (size_t)r * 1024 + tid * 4) = v;
}

__global__ void bias_sum_kernel(const float* __restrict__ a, const float* __restrict__ b,
                                float* __restrict__ o, int n)
{
    const int i = blockIdx.x * blockDim.x + threadIdx.x;
    if (i < n) o[i] = a[i] + b[i];
}

__device__ __forceinline__ float sigmoidf(float x) { return 1.0f / (1.0f + expf(-x)); }

// ---------------------------------------------------------------------------
// LSTM cell elementwise; scatters sel[b] at t == data_length[b]-1
// ---------------------------------------------------------------------------
__global__ void lstm_cell_kernel(const float* __restrict__ gates,
                                 float* __restrict__ h, float* __restrict__ c,
                                 float* __restrict__ sel,
                                 const int* __restrict__ data_length, int t)
{
    const int idx = blockIdx.x * blockDim.x + threadIdx.x;  // [0, 512*1024)
    const int b  = idx >> 10;
    const int jh = idx & 1023;
    const float* g = gates + (size_t)b * G4;
    const float gi = g[jh];
    const float gf = g[HID + jh];
    const float gg = g[2 * HID + jh];
    const float go = g[3 * HID + jh];
    const float cc = sigmoidf(gf) * c[idx] + sigmoidf(gi) * tanhf(gg);
    const float hh = sigmoidf(go) * tanhf(cc);
    c[idx] = cc;
    h[idx] = hh;
    if (t == data_length[b] - 1) sel[idx] = hh;
}

// ---------------------------------------------------------------------------
// out[b][o] = relu(b2[o] + dot(y1[b], W2[o]))   (512 x 2, trivial)
// ---------------------------------------------------------------------------
__global__ void policy_out_kernel(const float* __restrict__ y1, const float* __restrict__ W2,
                                  const float* __restrict__ b2, float* __restrict__ out)
{
    const int idx = blockIdx.x * blockDim.x + threadIdx.x;
    if (idx >= BATCH * 2) return;
    const int b = idx >> 1;
    const int o = idx & 1;
    const float* yb = y1 + (size_t)b * HID;
    const float* w  = W2 + (size_t)o * HID;
    float s = b2[o];
    for (int k = 0; k < HID; ++k) s = fmaf(yb[k], w[k], s);
    out[b * 2 + o] = fmaxf(s, 0.0f);
}

// ---------------------------------------------------------------------------
extern "C" void kernel_launch(void* const* d_in, const int* in_sizes, int n_in,
                              void* d_out, int out_size, void* d_ws, size_t ws_size,
                              hipStream_t stream)
{
    const int*   entity_batch   = (const int*)d_in[0];
    const int*   relation_batch = (const int*)d_in[1];
    // d_in[2] query_relations: unused by the reference
    const int*   data_length    = (const int*)d_in[3];
    const float* ent_emb        = (const float*)d_in[4];
    const float* rel_emb        = (const float*)d_in[5];
    const float* W_ih1          = (const float*)d_in[6] + (size_t)G4 * HID; // layer 1
    const float* W_hh1          = (const float*)d_in[7] + (size_t)G4 * HID; // layer 1
    const float* b_ih1          = (const float*)d_in[8] + G4;
    const float* b_hh1          = (const float*)d_in[9] + G4;
    const float* W1             = (const float*)d_in[10];
    const float* b1             = (const float*)d_in[11];
    const float* W2             = (const float*)d_in[12];
    const float* b2             = (const float*)d_in[13];
    float* out = (float*)d_out;

    // Workspace carve-up (~59 MB of f32)
    float* ws = (float*)d_ws;
    size_t off = 0;
    float* X     = ws + off; off += (size_t)BATCH * TSTEPS * HID; // [T*B][1024]
    float* bsum  = ws + off; off += G4;
    float* gates = ws + off; off += (size_t)BATCH * G4;           // [512][4096]
    float* h     = ws + off; off += (size_t)BATCH * HID;
    float* c     = ws + off; off += (size_t)BATCH * HID;
    float* sel   = ws + off; off += (size_t)BATCH * HID;
    float* y1    = ws + off; off += (size_t)BATCH * HID;
    (void)ws_size; (void)in_sizes; (void)n_in; (void)out_size;

    // 1) Fused bias, embedding gather
    bias_sum_kernel<<<G4 / 256, 256, 0, stream>>>(b_ih1, b_hh1, bsum, G4);
    build_x_kernel<<<BATCH * TSTEPS, 256, 0, stream>>>(entity_batch, relation_batch,
                                                       ent_emb, rel_emb, X);

    // 2) Zero h, c, sel (contiguous)
    hipMemsetAsync(h, 0, (size_t)3 * BATCH * HID * sizeof(float), stream);

    // 3) Recurrence: gates = h@W_hh1^T + x_t@W_ih1^T + bias, then LSTM cell
    for (int t = 0; t < TSTEPS; ++t) {
        gemm_wmma_f32<<<dim3(G4 / TN, BATCH / TM), 256, 0, stream>>>(
            h, W_hh1, HID,
            X + (size_t)t * BATCH * HID, W_ih1, HID,
            bsum, gates, G4, /*relu=*/0);
        lstm_cell_kernel<<<(BATCH * HID) / 256, 256, 0, stream>>>(
            gates, h, c, sel, data_length, t);
    }

    // 4) Policy MLP: y1 = relu(sel@W1^T + b1); out = relu(y1@W2^T + b2)
    gemm_wmma_f32<<<dim3(HID / TN, BATCH / TM), 256, 0, stream>>>(
        sel, W1, HID, nullptr, nullptr, 0, b1, y1, HID, /*relu=*/1);
    policy_out_kernel<<<(BATCH * 2 + 255) / 256, 256, 0, stream>>>(y1, W2, b2, out);
}